// TopologicalRegularizer_46746424049991
// MI455X (gfx1250) — compile-verified
//
#include <hip/hip_runtime.h>
#include <stdint.h>

#define BATCH 4
#define HH 384
#define WW 384
#define NPIX (HH * WW)        // 147456
#define NPAD (1 << 18)        // 262144  (next pow2 >= NPIX)

#define SHORT_T_C    0.08f
#define EXCESS_W_C   0.3f
#define SHORT_W_C    0.7f
#define LOSS_SCALE_C 10.0f
#define TARGET_B0_C  5.0f

typedef float v2f __attribute__((ext_vector_type(2)));
typedef float v8f __attribute__((ext_vector_type(8)));

// ---------------------------------------------------------------------------
// K1: f = 1 - prob*roi ; build u64 sort keys (f_bits<<32 | pix) ; init parent.
// Stages prob/roi through LDS with CDNA5 async global->LDS copies (ASYNCcnt).
// Grid: exactly BATCH*NPIX/4 threads, each handles 4 consecutive pixels (16B).
// ---------------------------------------------------------------------------
__global__ __launch_bounds__(256) void prep_kernel(
    const float* __restrict__ prob, const float* __restrict__ roi,
    float* __restrict__ f, unsigned long long* __restrict__ sortbuf,
    unsigned* __restrict__ parent)
{
    __shared__ __align__(16) unsigned char smem[256 * 32];

    const int t = blockIdx.x * 256 + threadIdx.x;       // quad index over BATCH*NPIX/4
    if (t * 4 >= BATCH * NPIX) return;

    const unsigned byteOff = (unsigned)t * 16u;          // 16 bytes per thread
    const unsigned lds_p = (unsigned)(size_t)(&smem[threadIdx.x * 32]);
    const unsigned lds_r = lds_p + 16u;
    const unsigned long long pbase = (unsigned long long)(size_t)prob;
    const unsigned long long rbase = (unsigned long long)(size_t)roi;

    // CDNA5 async copy: global -> LDS, 128b per lane, tracked by ASYNCcnt.
    asm volatile(
        "global_load_async_to_lds_b128 %0, %1, %2\n\t"
        "global_load_async_to_lds_b128 %3, %4, %5\n\t"
        "s_wait_asynccnt 0x0"
        :
        : "v"(lds_p), "v"(byteOff), "s"(pbase),
          "v"(lds_r), "v"(byteOff), "s"(rbase)
        : "memory");

    const float4 pv = *reinterpret_cast<const float4*>(&smem[threadIdx.x * 32]);
    const float4 rv = *reinterpret_cast<const float4*>(&smem[threadIdx.x * 32 + 16]);

    const unsigned idx0 = (unsigned)t * 4u;              // global pixel index (img*NPIX + pix)
    const unsigned img  = idx0 / NPIX;                   // NPIX % 4 == 0 -> all 4 in same image
    const unsigned pix0 = idx0 - img * NPIX;

    float fv[4];
    fv[0] = 1.0f - pv.x * rv.x;
    fv[1] = 1.0f - pv.y * rv.y;
    fv[2] = 1.0f - pv.z * rv.z;
    fv[3] = 1.0f - pv.w * rv.w;

#pragma unroll
    for (int k = 0; k < 4; ++k) {
        const unsigned pix = pix0 + (unsigned)k;
        f[idx0 + k] = fv[k];
        // f >= 0 so IEEE bit pattern is monotone; low 32 bits break ties by index.
        sortbuf[(size_t)img * NPAD + pix] =
            ((unsigned long long)__float_as_uint(fv[k]) << 32) | (unsigned long long)pix;
        parent[idx0 + k] = pix;
    }
}

// ---------------------------------------------------------------------------
// K2: fill sort padding with +inf keys so pads land at the end.
// ---------------------------------------------------------------------------
__global__ __launch_bounds__(256) void pad_kernel(unsigned long long* __restrict__ sortbuf)
{
    const int per = NPAD - NPIX;
    const int t = blockIdx.x * 256 + threadIdx.x;
    if (t >= BATCH * per) return;
    const int img = t / per;
    const int pos = NPIX + (t - img * per);
    sortbuf[(size_t)img * NPAD + pos] = ~0ull;
}

// ---------------------------------------------------------------------------
// K3a: LDS-resident bitonic phases. Each block owns 512 contiguous elements;
// partner i^j stays in-block for j<=256, so those steps run entirely in LDS.
// init variant: all k = 2..512.  final variant: j = 256..1 for a given k.
// Direction uses the GLOBAL element index: asc = ((base+i) & k) == 0.
// ---------------------------------------------------------------------------
__device__ __forceinline__ void local_bitonic_steps(
    unsigned long long* sh, unsigned base, int k, int jhi)
{
    for (int j = jhi; j > 0; j >>= 1) {
        __syncthreads();
        const unsigned t = threadIdx.x;
        const unsigned lo = t & (unsigned)(j - 1);
        const unsigned i = ((t ^ lo) << 1) | lo;     // comparator first element
        const unsigned p = i | (unsigned)j;          // partner
        const unsigned long long a = sh[i];
        const unsigned long long c = sh[p];
        const bool asc = (((base + i) & (unsigned)k) == 0);
        if (asc ? (a > c) : (a < c)) { sh[i] = c; sh[p] = a; }
    }
}

__global__ __launch_bounds__(256) void bitonic_local_init(unsigned long long* __restrict__ sortbuf)
{
    __shared__ unsigned long long sh[512];
    unsigned long long* b = sortbuf + (size_t)blockIdx.y * NPAD;
    const unsigned base = blockIdx.x * 512u;
    sh[threadIdx.x]       = b[base + threadIdx.x];
    sh[threadIdx.x + 256] = b[base + threadIdx.x + 256];
    for (int k = 2; k <= 512; k <<= 1)
        local_bitonic_steps(sh, base, k, k >> 1);
    __syncthreads();
    b[base + threadIdx.x]       = sh[threadIdx.x];
    b[base + threadIdx.x + 256] = sh[threadIdx.x + 256];
}

__global__ __launch_bounds__(256) void bitonic_local_final(
    unsigned long long* __restrict__ sortbuf, int k)
{
    __shared__ unsigned long long sh[512];
    unsigned long long* b = sortbuf + (size_t)blockIdx.y * NPAD;
    const unsigned base = blockIdx.x * 512u;
    sh[threadIdx.x]       = b[base + threadIdx.x];
    sh[threadIdx.x + 256] = b[base + threadIdx.x + 256];
    local_bitonic_steps(sh, base, k, 256);
    __syncthreads();
    b[base + threadIdx.x]       = sh[threadIdx.x];
    b[base + threadIdx.x + 256] = sh[threadIdx.x + 256];
}

// K3b: one global bitonic compare/exchange pass (j > 256). gridDim.y = image.
__global__ __launch_bounds__(256) void bitonic_step(
    unsigned long long* __restrict__ sortbuf, int j, int k)
{
    const int i = blockIdx.x * 256 + threadIdx.x;
    unsigned long long* b = sortbuf + (size_t)blockIdx.y * NPAD;
    const int l = i ^ j;
    if (l > i) {
        const unsigned long long a = b[i];
        const unsigned long long c = b[l];
        const bool asc = ((i & k) == 0);
        if (asc ? (a > c) : (a < c)) { b[i] = c; b[l] = a; }
    }
}

// ---------------------------------------------------------------------------
// K4: rank[pixel] = sorted position (first NPIX slots are the real entries).
// ---------------------------------------------------------------------------
__global__ __launch_bounds__(256) void rank_kernel(
    const unsigned long long* __restrict__ sortbuf, unsigned* __restrict__ rank)
{
    const int t = blockIdx.x * 256 + threadIdx.x;
    if (t >= BATCH * NPIX) return;
    const int img = t / NPIX;
    const int pos = t - img * NPIX;
    const unsigned pix = (unsigned)(sortbuf[(size_t)img * NPAD + pos] & 0xFFFFFFFFu);
    rank[(size_t)img * NPIX + pix] = (unsigned)pos;
}

// ---------------------------------------------------------------------------
// K5: serial elder-rule union-find per image (one thread each, 4 concurrent).
// Vertex-ordered processing == edge Kruskal with edge value = max endpoint.
// ---------------------------------------------------------------------------
__device__ __forceinline__ unsigned uf_find(unsigned* __restrict__ par, unsigned x)
{
    unsigned root = x, p;
    while ((p = par[root]) != root) root = p;
    while (par[x] != root) { unsigned nxt = par[x]; par[x] = root; x = nxt; }
    return root;
}

__global__ void uf_kernel(const float* __restrict__ f,
                          const unsigned long long* __restrict__ sortbuf,
                          const unsigned* __restrict__ rank,
                          unsigned* __restrict__ parent,
                          float* __restrict__ losses)
{
    if (threadIdx.x != 0) return;
    const int img = blockIdx.x;
    const float* fi = f + (size_t)img * NPIX;
    const unsigned* rk = rank + (size_t)img * NPIX;
    unsigned* par = parent + (size_t)img * NPIX;
    const unsigned long long* sb = sortbuf + (size_t)img * NPAD;

    long long bars = 0;
    double spen = 0.0;

    for (int pos = 0; pos < NPIX; ++pos) {
        const unsigned long long e = sb[pos];
        const unsigned v = (unsigned)(e & 0xFFFFFFFFu);
        const float fv = __uint_as_float((unsigned)(e >> 32));   // key == f(v)

        // Lookahead prefetch of a future vertex's f/rank lines (serial loop is
        // pure L2-latency; the sorted stream gives exact future addresses).
        if (pos + 6 < NPIX) {
            const unsigned vn = (unsigned)(sb[pos + 6] & 0xFFFFFFFFu);
            __builtin_prefetch(&fi[vn], 0, 0);
            __builtin_prefetch(&rk[vn], 0, 0);
        }

        const int x = (int)(v % WW);
        const int y = (int)(v / WW);
        unsigned rcur = v;                       // v is its own root when added

        unsigned nb[4]; int nn = 0;
        if (x > 0)      nb[nn++] = v - 1;
        if (x < WW - 1) nb[nn++] = v + 1;
        if (y > 0)      nb[nn++] = v - WW;
        if (y < HH - 1) nb[nn++] = v + WW;

        for (int q = 0; q < nn; ++q) {
            const unsigned u = nb[q];
            if (rk[u] < (unsigned)pos) {          // neighbor already in filtration
                unsigned ru = uf_find(par, u);
                unsigned rv = uf_find(par, rcur);
                if (ru == rv) { rcur = rv; continue; }
                const float bu = fi[ru];
                const float bv = fi[rv];
                unsigned live, die; float bdie;
                if (bu <= bv) { live = ru; die = rv; bdie = bv; }   // elder rule
                else          { live = rv; die = ru; bdie = bu; }
                if (fv > bdie) {                  // positive persistence only
                    ++bars;
                    const float life = fv - bdie;
                    const float s = SHORT_T_C - life;
                    if (s > 0.0f) spen += (double)s;
                }
                par[die] = live;
                rcur = live;
            }
        }
    }

    float excess = (float)bars - TARGET_B0_C;
    if (excess < 0.0f) excess = 0.0f;
    losses[img] = excess * excess * EXCESS_W_C + (float)spen * SHORT_W_C;
}

// ---------------------------------------------------------------------------
// K6: batch reduction on the matrix pipe: D = A x B with A row 0 = losses,
// B column 0 = ones  =>  D[0,0] = sum(losses).  One full wave32, EXEC all-1s.
// A (16x4 f32): lane m<16 holds {K0,K1}, lane m+16 holds {K2,K3}.
// B (4x16 f32): vgpr0 = {K0 | K2} rows, vgpr1 = {K1 | K3} rows, lanes = N.
// ---------------------------------------------------------------------------
__global__ __launch_bounds__(32) void final_kernel(
    const float* __restrict__ losses, float* __restrict__ out)
{
    const unsigned lane = threadIdx.x;
    v2f a = {0.0f, 0.0f};
    v2f b = {0.0f, 0.0f};
    if (lane == 0)  { a[0] = losses[0]; a[1] = losses[1]; b[0] = 1.0f; b[1] = 1.0f; }
    if (lane == 16) { a[0] = losses[2]; a[1] = losses[3]; b[0] = 1.0f; b[1] = 1.0f; }
    v8f c = {};
    c = __builtin_amdgcn_wmma_f32_16x16x4_f32(
            /*neg_a=*/false, a, /*neg_b=*/false, b,
            /*c_mod=*/(short)0, c, /*reuse_a=*/false, /*reuse_b=*/false);
    if (lane == 0)
        out[0] = c[0] * (LOSS_SCALE_C / (float)BATCH);   // mean * 10
}

// ---------------------------------------------------------------------------
extern "C" void kernel_launch(void* const* d_in, const int* in_sizes, int n_in,
                              void* d_out, int out_size, void* d_ws, size_t ws_size,
                              hipStream_t stream)
{
    const float* prob = (const float*)d_in[0];
    const float* roi  = (const float*)d_in[1];
    float* out = (float*)d_out;

    // Workspace layout (all 16B aligned):
    char* ws = (char*)d_ws;
    const size_t f_bytes    = (size_t)BATCH * NPIX * sizeof(float);          // 2.25 MiB
    const size_t rank_bytes = (size_t)BATCH * NPIX * sizeof(unsigned);       // 2.25 MiB
    const size_t par_bytes  = (size_t)BATCH * NPIX * sizeof(unsigned);       // 2.25 MiB
    float*              f_buf   = (float*)(ws);
    unsigned*           rank_b  = (unsigned*)(ws + f_bytes);
    unsigned*           par_b   = (unsigned*)(ws + f_bytes + rank_bytes);
    float*              loss_b  = (float*)(ws + f_bytes + rank_bytes + par_bytes);
    unsigned long long* sort_b  = (unsigned long long*)(ws + f_bytes + rank_bytes + par_bytes + 64);

    // K1: f/key/parent init (exactly BATCH*NPIX/4 threads)
    {
        const int threads = BATCH * NPIX / 4;            // 147456
        prep_kernel<<<dim3((threads + 255) / 256), dim3(256), 0, stream>>>(
            prob, roi, f_buf, sort_b, par_b);
    }
    // K2: pad tail of each image's sort segment
    {
        const int n = BATCH * (NPAD - NPIX);
        pad_kernel<<<dim3((n + 255) / 256), dim3(256), 0, stream>>>(sort_b);
    }
    // K3: bitonic sort; j<=256 phases fused into LDS-resident kernels.
    {
        const dim3 lgrid(NPAD / 512, BATCH);
        const dim3 ggrid(NPAD / 256, BATCH);
        bitonic_local_init<<<lgrid, dim3(256), 0, stream>>>(sort_b);
        for (int k = 1024; k <= NPAD; k <<= 1) {
            for (int j = k >> 1; j > 256; j >>= 1)
                bitonic_step<<<ggrid, dim3(256), 0, stream>>>(sort_b, j, k);
            bitonic_local_final<<<lgrid, dim3(256), 0, stream>>>(sort_b, k);
        }
    }
    // K4: rank scatter
    {
        const int n = BATCH * NPIX;
        rank_kernel<<<dim3((n + 255) / 256), dim3(256), 0, stream>>>(sort_b, rank_b);
    }
    // K5: serial union-find, one image per block
    uf_kernel<<<dim3(BATCH), dim3(32), 0, stream>>>(f_buf, sort_b, rank_b, par_b, loss_b);

    // K6: final reduction via V_WMMA_F32_16X16X4_F32
    final_kernel<<<dim3(1), dim3(32), 0, stream>>>(loss_b, out);
}